// VNETDetector_69638599737924
// MI455X (gfx1250) — compile-verified
//
#include <hip/hip_runtime.h>
#include <hip/hip_bf16.h>
#include <stdint.h>

// ---------------------------------------------------------------------------
// VNET detector for MI455X (gfx1250, wave32).
//   Phase 0: MLP  out = relu(relu(rx*W1+b1)@W2+b2)@W3+b3, store llr = -out.
//            Uses V_WMMA_F32_16X16X4_F32 (f32 WMMA keeps reference precision).
//   Phase A: per-block 32x32 min-plus (tropical) transfer matrices.
//   Phase B: sequential chain of 512 min-plus matvecs (one wave).
//   Phase C: parallel per-block replay, emit bits.
// The trellis tt[i]=[2i%64,(2i+1)%64] makes v[i]==v[i+32] after every step,
// so the Viterbi state is exactly 32 floats = one wave32 lane each.
// ---------------------------------------------------------------------------

#define T_LEN   131072
#define NB      512      // number of scan blocks
#define LBLK    256      // steps per block (NB*LBLK == T_LEN)
#define BIGF    1e30f    // tropical "infinity"

typedef float v2f __attribute__((ext_vector_type(2)));
typedef float v8f __attribute__((ext_vector_type(8)));

// ---------------------------------------------------------------------------
// Phase 0: MLP via f32 WMMA. One wave handles 16 timesteps.
//   layer2: C2[16][32] = h1[16][64] x W2[64][32]   (16 K-steps, 2 N-tiles)
//   layer3: C3[16][64] = h2[16][32] x W3[32][64]   ( 8 K-steps, 4 N-tiles)
// h2 is transposed C-layout -> A-layout through a per-wave LDS slice.
// ---------------------------------------------------------------------------
__launch_bounds__(128)
__global__ void vnet_mlp_kernel(const float* __restrict__ rx,
                                const float* __restrict__ W1,
                                const float* __restrict__ b1,
                                const float* __restrict__ W2,
                                const float* __restrict__ b2,
                                const float* __restrict__ W3,
                                const float* __restrict__ b3,
                                float* __restrict__ llr) {
    __shared__ float h2s[4][16 * 32];   // 2 KB per wave

    const int wave  = threadIdx.x >> 5;
    const int lane  = threadIdx.x & 31;
    const int m     = lane & 15;        // row (timestep) within tile
    const int hi    = lane >> 4;        // half-wave selector
    const int tile  = blockIdx.x * 4 + wave;
    const int t0    = tile * 16;

    const float rxv = rx[t0 + m];       // rx[t,0]

    // ---- layer 2: h1 @ W2, two 16-wide N tiles -----------------------------
    v8f acc2a = {}; v8f acc2b = {};
    #pragma unroll
    for (int kb = 0; kb < 16; ++kb) {
        const int k0 = kb * 4 + 2 * hi;
        v2f a;
        a.x = fmaxf(rxv * W1[k0]     + b1[k0],     0.0f);   // h1[m][k0]
        a.y = fmaxf(rxv * W1[k0 + 1] + b1[k0 + 1], 0.0f);   // h1[m][k0+1]
        v2f bA, bB;
        bA.x = W2[(k0    ) * 32 +  m      ];
        bA.y = W2[(k0 + 1) * 32 +  m      ];
        bB.x = W2[(k0    ) * 32 + (m + 16)];
        bB.y = W2[(k0 + 1) * 32 + (m + 16)];
        acc2a = __builtin_amdgcn_wmma_f32_16x16x4_f32(false, a, false, bA,
                                                      (short)0, acc2a, false, false);
        acc2b = __builtin_amdgcn_wmma_f32_16x16x4_f32(false, a, false, bB,
                                                      (short)0, acc2b, false, false);
    }

    // bias + relu, stage h2 tile (16x32) into this wave's LDS slice
    {
        const float b2a = b2[m];
        const float b2b = b2[m + 16];
        #pragma unroll
        for (int v = 0; v < 8; ++v) {
            const int row = v + 8 * hi;
            h2s[wave][row * 32 + m]      = fmaxf(acc2a[v] + b2a, 0.0f);
            h2s[wave][row * 32 + m + 16] = fmaxf(acc2b[v] + b2b, 0.0f);
        }
    }
    __syncthreads();

    // ---- layer 3: h2 @ W3, four 16-wide N tiles ----------------------------
    v8f acc3[4] = {{}, {}, {}, {}};
    #pragma unroll
    for (int kb = 0; kb < 8; ++kb) {
        const int k0 = kb * 4 + 2 * hi;
        v2f a;
        a.x = h2s[wave][m * 32 + k0];
        a.y = h2s[wave][m * 32 + k0 + 1];
        #pragma unroll
        for (int j = 0; j < 4; ++j) {
            const int n0 = j * 16;
            v2f b;
            b.x = W3[(k0    ) * 64 + n0 + m];
            b.y = W3[(k0 + 1) * 64 + n0 + m];
            acc3[j] = __builtin_amdgcn_wmma_f32_16x16x4_f32(false, a, false, b,
                                                            (short)0, acc3[j],
                                                            false, false);
        }
    }

    // epilogue: llr[t][n] = -(C3 + b3)
    #pragma unroll
    for (int j = 0; j < 4; ++j) {
        const int   n0  = j * 16;
        const float b3v = b3[n0 + m];
        #pragma unroll
        for (int v = 0; v < 8; ++v) {
            const int t = t0 + v + 8 * hi;
            llr[(size_t)t * 64 + n0 + m] = -(acc3[j][v] + b3v);
        }
    }
}

// ---------------------------------------------------------------------------
// Phase A: per-block 32x32 tropical transfer matrix. One wave per block;
// lane j owns column j, so the update is purely lane-local VALU and the llr
// reads are wave-uniform (scalar loads).
//   M'[i][j] = min(M[(2i)%32][j] + llr[2i],  M[(2i+1)%32][j] + llr[2i+1])
// Stored row-major: mats[b][i][j].
// ---------------------------------------------------------------------------
__launch_bounds__(256)
__global__ void vnet_blockmat_kernel(const float* __restrict__ llr,
                                     float* __restrict__ mats) {
    const int b    = (blockIdx.x * blockDim.x + threadIdx.x) >> 5;
    const int lane = threadIdx.x & 31;

    float col[32];
    #pragma unroll
    for (int i = 0; i < 32; ++i) col[i] = (i == lane) ? 0.0f : BIGF;

    const float* lp = llr + (size_t)b * LBLK * 64;
    for (int s = 0; s < LBLK; ++s) {
        float nc[32];
        #pragma unroll
        for (int i = 0; i < 32; ++i) {
            const float l0 = lp[2 * i];       // uniform -> s_load
            const float l1 = lp[2 * i + 1];
            nc[i] = fminf(col[(2 * i) & 31] + l0,
                          col[(2 * i + 1) & 31] + l1);
        }
        #pragma unroll
        for (int i = 0; i < 32; ++i) col[i] = nc[i];
        lp += 64;
    }

    float* mb = mats + (size_t)b * 1024 + lane;   // [i][j=lane], coalesced
    #pragma unroll
    for (int i = 0; i < 32; ++i) mb[i * 32] = col[i];
}

// ---------------------------------------------------------------------------
// Phase B: sequential min-plus matvec chain; single wave. Lane i holds row i.
//   u_{b+1}[i] = min_j ( M_b[i][j] + u_b[j] )
// Records u at every block boundary into bvec.
// ---------------------------------------------------------------------------
__launch_bounds__(32)
__global__ void vnet_scan_kernel(const float* __restrict__ mats,
                                 float* __restrict__ bvec) {
    const int lane = threadIdx.x;   // row index
    float u = 0.0f;                 // u_0 = zeros
    for (int b = 0; b < NB; ++b) {
        bvec[b * 32 + lane] = u;
        const float* row = mats + (size_t)b * 1024 + lane * 32;
        float r = BIGF;
        #pragma unroll
        for (int j = 0; j < 32; ++j) {
            const float uj = __shfl(u, j, 32);
            r = fminf(r, row[j] + uj);
        }
        u = r;
    }
}

// ---------------------------------------------------------------------------
// Phase C: parallel replay. One wave per block, lane i = state i.
//   bit_t = (index of first minimum of u) & 1, then update u.
// ---------------------------------------------------------------------------
__launch_bounds__(256)
__global__ void vnet_emit_kernel(const float* __restrict__ llr,
                                 const float* __restrict__ bvec,
                                 float* __restrict__ out) {
    const int b    = (blockIdx.x * blockDim.x + threadIdx.x) >> 5;
    const int lane = threadIdx.x & 31;

    float u = bvec[b * 32 + lane];
    const float* lp = llr + (size_t)b * LBLK * 64;
    float*       op = out + (size_t)b * LBLK;

    for (int s = 0; s < LBLK; ++s) {
        // wave-wide min (butterfly), then first-occurrence argmin via ballot
        float mn = u;
        #pragma unroll
        for (int off = 16; off > 0; off >>= 1)
            mn = fminf(mn, __shfl_xor(mn, off, 32));
        const unsigned long long mask = __ballot(u == mn);
        const int idx = __ffsll(mask) - 1;          // first minimum lane (<32)
        if (lane == 0) op[s] = (float)(idx & 1);

        // u'[i] = min(u[(2i)%32] + llr[2i], u[(2i+1)%32] + llr[2i+1])
        const float l0 = lp[2 * lane];
        const float l1 = lp[2 * lane + 1];
        const float p0 = __shfl(u, (2 * lane) & 31, 32) + l0;
        const float p1 = __shfl(u, (2 * lane + 1) & 31, 32) + l1;
        u = fminf(p0, p1);
        lp += 64;
    }
}

// ---------------------------------------------------------------------------
// Host entry. Inputs: rx, W1, b1, W2, b2, W3, b3, n_states.
// Workspace: llr[T*64] | mats[NB*1024] | bvec[NB*32]  (~34.3 MB, fits L2).
// ---------------------------------------------------------------------------
extern "C" void kernel_launch(void* const* d_in, const int* in_sizes, int n_in,
                              void* d_out, int out_size, void* d_ws, size_t ws_size,
                              hipStream_t stream) {
    const float* rx = (const float*)d_in[0];
    const float* W1 = (const float*)d_in[1];
    const float* b1 = (const float*)d_in[2];
    const float* W2 = (const float*)d_in[3];
    const float* b2 = (const float*)d_in[4];
    const float* W3 = (const float*)d_in[5];
    const float* b3 = (const float*)d_in[6];
    float* out = (float*)d_out;

    float* llr  = (float*)d_ws;                       // T*64 floats
    float* mats = llr  + (size_t)T_LEN * 64;          // NB*1024 floats
    float* bvec = mats + (size_t)NB * 1024;           // NB*32 floats

    // Phase 0: MLP. 8192 tiles of 16 timesteps, 4 waves per 128-thread block.
    vnet_mlp_kernel<<<T_LEN / 16 / 4, 128, 0, stream>>>(rx, W1, b1, W2, b2,
                                                        W3, b3, llr);
    // Phase A: 512 block matrices, 8 waves per 256-thread block.
    vnet_blockmat_kernel<<<NB / 8, 256, 0, stream>>>(llr, mats);
    // Phase B: single-wave sequential chain over 512 blocks.
    vnet_scan_kernel<<<1, 32, 0, stream>>>(mats, bvec);
    // Phase C: parallel replay + bit emission.
    vnet_emit_kernel<<<NB / 8, 256, 0, stream>>>(llr, bvec, out);
}